// FourierLayer_56616258896283
// MI455X (gfx1250) — compile-verified
//
#include <hip/hip_runtime.h>

// ============================================================================
// FNO spectral layer for MI455X (gfx1250), implemented as a chain of partial
// DFT GEMMs on V_WMMA_F32_16X16X4_F32 (fp32 matrix pipe; fp16/bf16 twiddles
// would destroy spectral precision). No full FFT is ever computed: only 32
// modes per axis are kept, so each axis transform is a dense matrix
// contraction. Twiddle (B) panels are staged into LDS (pre-paired into
// float2 fragments -> one ds_load_b64 per fragment) so the inner loop's
// vector-memory slots are reserved for the streaming A operand.
//
// Shapes: B=2, C=32, T=64, X=128, Y=128, 32 modes per axis.
// Roofline: ~40 GFLOP fp32; ~0.8 GB mandatory HBM traffic + ~1 GB of
// intermediate mode-plane traffic (largely L2-resident, 192 MB L2).
// Workspace requirement: ~192.2 MB (layout in kernel_launch).
// ============================================================================

typedef float v2f __attribute__((ext_vector_type(2)));
typedef float v8f __attribute__((ext_vector_type(8)));

__device__ __forceinline__ v8f wmma_f32_4(v2f a, v2f b, v8f c) {
#if __has_builtin(__builtin_amdgcn_wmma_f32_16x16x4_f32)
  // 8-arg f32 pattern: (neg_a, A, neg_b, B, c_mod, C, reuse_a, reuse_b)
  return __builtin_amdgcn_wmma_f32_16x16x4_f32(false, a, false, b, (short)0, c,
                                               false, false);
#else
  // Compile-survival fallback only (should not be taken on gfx1250).
  c[0] += a.x * b.x;
  return c;
#endif
}

// ----------------------------------------------------------------------------
// Twiddle-matrix builder. Each twiddle set is 3 contiguous K*N planes:
//   [Bre | Bim | BimNeg]   (BimNeg = -Bim, so the GEMM kernel never needs a
//   negate modifier — the ISA NEG field on f32 WMMA is C-negate only).
// Forward DFT:  w = e^{-i th} -> Bre=cos, Bim=-sin.  Inverse: w = e^{+i th}.
// kmap: 0 = identity (time modes)
//       1 = forward Y map  : f = m<16 ? m+49  : m-16      (fftshift of rfft65)
//       2 = X map (fwd&inv): f = m<16 ? m+112 : m-16      (even axis)
//       3 = inverse Y map  : f = m<17 ? m+48  : m-17      (double fftshift on
//                                                          odd axis = shift -1)
// finalScale: fold Hermitian doubling (c=2 except ky in {0,64}) and the
//             1/(64*128*128) irfftn normalization into the last stage.
// ----------------------------------------------------------------------------
__global__ __launch_bounds__(256) void build_twiddle(
    float* __restrict__ dst, int K, int N, int period, int kmap, float sign,
    int modeIsCol, int finalScale) {
  int idx = blockIdx.x * blockDim.x + threadIdx.x;
  int total = K * N;
  if (idx >= total) return;
  int k = idx / N, n = idx % N;
  int modeIdx = modeIsCol ? n : k;
  int spat = modeIsCol ? k : n;
  int f;
  if (kmap == 0)      f = modeIdx;
  else if (kmap == 1) f = (modeIdx < 16) ? modeIdx + 49 : modeIdx - 16;
  else if (kmap == 2) f = (modeIdx < 16) ? modeIdx + 112 : modeIdx - 16;
  else                f = (modeIdx < 17) ? modeIdx + 48 : modeIdx - 17;
  int phase = (f * spat) % period;  // integer phase reduction for accuracy
  float theta = 6.283185307179586f * (float)phase / (float)period;
  float wr = cosf(theta);
  float wi = sign * sinf(theta);
  float scale = 1.0f;
  if (finalScale) {
    float c = (f == 0 || f == 64) ? 1.0f : 2.0f;  // DC / Nyquist counted once
    scale = c * (1.0f / 1048576.0f);              // 1/(64*128*128)
  }
  dst[idx] = scale * wr;
  dst[total + idx] = scale * wi;
  dst[2 * total + idx] = -scale * wi;
}

// ----------------------------------------------------------------------------
// Generic strided complex GEMM on WMMA f32 16x16x4.
//   C[row, n] (+)= sum_k A[row, k] * W[k, n]          (complex)
// A addressing: row = o*D + m  ->  elem = aRe/aIm[ o*S1 + m*S2 + k*Sk ]
// W (twiddle)  : dense [K x N] planes Bre/Bim/BimNeg, staged to LDS with the
//                two K-rows of each fragment pre-paired into a float2 so a
//                fragment read is a single ds_load_b64.
// C            : dense row-major [M x N] (separate re/im planes).
// One wave computes a 16(M) x 32(N) tile: 2 accumulator pairs, K-step 4.
// Wave32 fragment layout per ISA 7.12.2:
//   A 16x4 : lane<16 -> row=lane,    {K=k0,k0+1}; lane>=16 -> {K=k0+2,k0+3}
//   B 4x16 : lane<16 -> col=lane,    {K=k0,k0+1}; lane>=16 -> {K=k0+2,k0+3}
//   C/D    : lane<16 -> col=lane, rows 0..7 in v0..7; lane>=16 -> rows 8..15
// EXEC is all-ones inside the compute path (uniform tile-bound branch only,
// placed after the barrier).
// ----------------------------------------------------------------------------
template <bool HAS_IMAG, bool WRITE_IMAG>
__global__ __launch_bounds__(256) void cgemm_wmma(
    const float* __restrict__ aRe, const float* __restrict__ aIm,
    const float* __restrict__ tw, float* __restrict__ cRe,
    float* __restrict__ cIm, int M, int N, int K, int D, int S1, int S2,
    int Sk) {
  // 48 KB: 3 planes of (K/2)*N float2 fragments (max K*N = 4096).
  __shared__ v2f smem[6144];

  const int tid = threadIdx.x;
  const int KN2 = (K >> 1) * N;  // fragments per plane
  for (int i = tid; i < 3 * KN2; i += 256) {
    const int plane = i / KN2;
    const int idx = i - plane * KN2;
    const int kk = idx / N;        // K-pair index (rows 2kk, 2kk+1)
    const int col = idx - kk * N;
    const float* src = tw + (size_t)plane * K * N + (size_t)(2 * kk) * N + col;
    v2f v;
    v.x = src[0];
    v.y = src[N];
    smem[i] = v;
  }
  __syncthreads();

  const int lane = tid & 31;
  const int wave = tid >> 5;
  const int tile = blockIdx.x * 8 + wave;
  const int ntilesN = N >> 5;  // 32-wide N tiles
  const int tileM = tile / ntilesN;
  const int tileN = tile % ntilesN;
  if (tileM * 16 >= M) return;

  const int l15 = lane & 15;
  const int hi = lane >> 4;

  const int r = tileM * 16 + l15;
  const long aBase = (long)(r / D) * S1 + (long)(r % D) * S2;

  const v2f* __restrict__ bRe = smem;
  const v2f* __restrict__ bIm = smem + KN2;
  const v2f* __restrict__ bImN = smem + 2 * KN2;
  const int col0 = tileN * 32;
  const int c0 = col0 + l15;
  const int c1 = col0 + 16 + l15;

  v8f accRe0 = {0, 0, 0, 0, 0, 0, 0, 0};
  v8f accRe1 = accRe0, accIm0 = accRe0, accIm1 = accRe0;

  for (int k0 = 0; k0 < K; k0 += 4) {
    const int ka = k0 + hi * 2;
    v2f aR, aI = {0.f, 0.f};
    aR.x = aRe[aBase + (long)ka * Sk];
    aR.y = aRe[aBase + (long)(ka + 1) * Sk];
    if (HAS_IMAG) {
      aI.x = aIm[aBase + (long)ka * Sk];
      aI.y = aIm[aBase + (long)(ka + 1) * Sk];
    }
    // prefetch next A K-panel (speculative; dropped if OOB per ISA)
    __builtin_prefetch(&aRe[aBase + (long)(ka + 4) * Sk], 0, 0);

    const int pr = (k0 >> 1) + hi;  // fragment (K-pair) row in LDS panel
    const int f0 = pr * N + c0;
    const int f1 = pr * N + c1;

    {  // N-subtile 0
      const v2f bR = bRe[f0];
      accRe0 = wmma_f32_4(aR, bR, accRe0);
      if (HAS_IMAG) accRe0 = wmma_f32_4(aI, bImN[f0], accRe0);
      if (WRITE_IMAG) {
        accIm0 = wmma_f32_4(aR, bIm[f0], accIm0);
        if (HAS_IMAG) accIm0 = wmma_f32_4(aI, bR, accIm0);
      }
    }
    {  // N-subtile 1
      const v2f bR = bRe[f1];
      accRe1 = wmma_f32_4(aR, bR, accRe1);
      if (HAS_IMAG) accRe1 = wmma_f32_4(aI, bImN[f1], accRe1);
      if (WRITE_IMAG) {
        accIm1 = wmma_f32_4(aR, bIm[f1], accIm1);
        if (HAS_IMAG) accIm1 = wmma_f32_4(aI, bR, accIm1);
      }
    }
  }

  const int rowBase = tileM * 16 + hi * 8;
#pragma unroll
  for (int q = 0; q < 8; ++q) {
    const long off = (long)(rowBase + q) * N;
    cRe[off + c0] = accRe0[q];
    cRe[off + c1] = accRe1[q];
    if (WRITE_IMAG) {
      cIm[off + c0] = accIm0[q];
      cIm[off + c1] = accIm1[q];
    }
  }
}

// ----------------------------------------------------------------------------
// Mode mixing: out[b,c,j,p,t] = sum_i S[b,c,j,p,i] * W[i, c, t, p, j]
// (the reference einsum contracts the TIME-mode axis with the weights' first
// axis; channel passes through).  ~0.5 GFLOP, weight-traffic bound -> VALU.
// S / out layout: [b, c, j(y-mode), p(x-mode), tmode], contiguous.
// ----------------------------------------------------------------------------
__global__ __launch_bounds__(256) void mix_modes(
    const float* __restrict__ Sre, const float* __restrict__ Sim,
    const float* __restrict__ wRe, const float* __restrict__ wIm,
    float* __restrict__ Ore, float* __restrict__ Oim) {
  int tid = blockIdx.x * blockDim.x + threadIdx.x;  // 2*32*32*32*32 threads
  int t = tid & 31;
  int p = (tid >> 5) & 31;
  int j = (tid >> 10) & 31;
  int c = (tid >> 15) & 31;
  int b = tid >> 20;
  size_t sbase = ((((size_t)((b * 32 + c) * 32 + j)) * 32 + p) * 32);
  size_t wbase = (size_t)c * 32768 + (size_t)t * 1024 + (size_t)p * 32 + j;
  float orr = 0.f, oii = 0.f;
#pragma unroll 4
  for (int i = 0; i < 32; ++i) {
    float sr = Sre[sbase + i], si = Sim[sbase + i];
    float wr = wRe[wbase + (size_t)i * 1048576];
    float wi = wIm[wbase + (size_t)i * 1048576];
    orr = fmaf(sr, wr, fmaf(-si, wi, orr));
    oii = fmaf(sr, wi, fmaf(si, wr, oii));
  }
  Ore[sbase + t] = orr;
  Oim[sbase + t] = oii;
}

// ============================================================================
extern "C" void kernel_launch(void* const* d_in, const int* in_sizes, int n_in,
                              void* d_out, int out_size, void* d_ws,
                              size_t ws_size, hipStream_t stream) {
  const float* x = (const float*)d_in[0];       // [2,32,64,128,128]
  const float* w_real = (const float*)d_in[1];  // [32,32,32,32,32]
  const float* w_imag = (const float*)d_in[2];
  float* out = (float*)d_out;  // [2,32,64,128,128]
  float* ws = (float*)d_ws;

  // ---- workspace layout (floats). Total = 50,397,184 floats ~= 192.2 MB ----
  float* twA = ws + 0;      // Y-fwd:  K=128,N=32 (3 planes = 12288)
  float* twB = ws + 12288;  // X-fwd:  K=128,N=32
  float* twC = ws + 24576;  // T-fwd:  K=64, N=32 (6144)
  float* twD = ws + 30720;  // T-inv:  K=32, N=64 (6144)
  float* twE = ws + 36864;  // X-inv:  K=32, N=128 (12288)
  float* twF = ws + 49152;  // Y-inv:  K=32, N=128 (12288)
  float* Gre = ws + 65536;  // [b,c,t,x, j] 16.78M floats per plane
  float* Gim = Gre + 16777216;
  float* Hre = ws + 65536 + 33554432;  // [b,c,t,j,p] 4.19M per plane
  float* Him = Hre + 4194304;
  float* Sre = Hre + 8388608;  // [b,c,j,p,tm] 2.10M per plane
  float* Sim = Sre + 2097152;
  float* S2re = Sre + 4194304;  // mixed modes, same layout
  float* S2im = S2re + 2097152;
  float* Ure = Hre;  // [b,c,j,p,t64]  (reuses H region, free after T-fwd)
  float* Uim = Ure + 4194304;
  float* Vre = Gre;  // [b,c,j,t,x]    (reuses G region, free after X-fwd)
  float* Vim = Vre + 16777216;

  // ---- twiddle matrices ----
  build_twiddle<<<(128 * 32 + 255) / 256, 256, 0, stream>>>(twA, 128, 32, 128,
                                                            1, -1.f, 1, 0);
  build_twiddle<<<(128 * 32 + 255) / 256, 256, 0, stream>>>(twB, 128, 32, 128,
                                                            2, -1.f, 1, 0);
  build_twiddle<<<(64 * 32 + 255) / 256, 256, 0, stream>>>(twC, 64, 32, 64, 0,
                                                           -1.f, 1, 0);
  build_twiddle<<<(32 * 64 + 255) / 256, 256, 0, stream>>>(twD, 32, 64, 64, 0,
                                                           +1.f, 0, 0);
  build_twiddle<<<(32 * 128 + 255) / 256, 256, 0, stream>>>(twE, 32, 128, 128,
                                                            2, +1.f, 0, 0);
  build_twiddle<<<(32 * 128 + 255) / 256, 256, 0, stream>>>(twF, 32, 128, 128,
                                                            3, +1.f, 0, 1);

  // ---- forward partial DFTs ----
  // Y: rows=(b,c,t,x)=524288, K=128(y), N=32(j). x real input.
  cgemm_wmma<false, true><<<4096, 256, 0, stream>>>(
      x, nullptr, twA, Gre, Gim, 524288, 32, 128, 128, 16384, 128, 1);
  // X: rows=(b,c,t,j)=131072, K=128(x) stride 32 inside G, N=32(p).
  cgemm_wmma<true, true><<<1024, 256, 0, stream>>>(
      Gre, Gim, twB, Hre, Him, 131072, 32, 128, 32, 4096, 1, 32);
  // T: rows=(b,c,j,p)=65536, K=64(t) stride 1024 inside H, N=32(tm).
  cgemm_wmma<true, true><<<512, 256, 0, stream>>>(
      Hre, Him, twC, Sre, Sim, 65536, 32, 64, 1024, 65536, 1, 1024);

  // ---- complex mode mixing (einsum) ----
  mix_modes<<<8192, 256, 0, stream>>>(Sre, Sim, w_real, w_imag, S2re, S2im);

  // ---- inverse partial DFTs ----
  // T-inv: rows=(b,c,j,p)=65536, K=32(tm), N=64(t).
  cgemm_wmma<true, true><<<1024, 256, 0, stream>>>(
      S2re, S2im, twD, Ure, Uim, 65536, 64, 32, 1, 32, 0, 1);
  // X-inv: rows=(b,c,j,t)=131072, K=32(p) stride 64 inside U, N=128(x).
  cgemm_wmma<true, true><<<4096, 256, 0, stream>>>(
      Ure, Uim, twE, Vre, Vim, 131072, 128, 32, 64, 2048, 1, 64);
  // Y-inv (real output): rows=(b,c,t,x)=524288, K=32(j) stride 8192 in V,
  // N=128(y); Hermitian doubling + 1/(T*X*Y) folded into twF.
  cgemm_wmma<true, false><<<16384, 256, 0, stream>>>(
      Vre, Vim, twF, out, nullptr, 524288, 128, 32, 8192, 262144, 1, 8192);
}